// AttentionLSTMAutoencoder_40441412059866
// MI455X (gfx1250) — compile-verified
//
#include <hip/hip_runtime.h>
#include <math.h>

// ---------------------------------------------------------------------------
// AttentionLSTMAutoencoder for MI455X (gfx1250, wave32).
// - Input projections xp = x*Wih^T + bih + bhh run as full-chip parallel
//   WMMA GEMMs (V_WMMA_F32_16X16X4_F32), off the recurrent critical path.
// - LSTM scans are persistent single-WGP kernels: h lives in LDS, cell state
//   c lives in per-wave registers across all T steps; gate accumulators are
//   initialized from the precomputed xp tile (C-matrix layout loads).
// - Falls back to a fused (xp-in-scan) path if the workspace is too small.
// ---------------------------------------------------------------------------

typedef float v2f __attribute__((ext_vector_type(2)));
typedef float v8f __attribute__((ext_vector_type(8)));

#define B_  128
#define T_  128
#define D_  64
#define HA_ 32

__device__ __forceinline__ float sigm(float x) { return 1.0f / (1.0f + __expf(-x)); }

__device__ __forceinline__ v8f vzero8() {
    v8f v;
#pragma unroll
    for (int i = 0; i < 8; ++i) v[i] = 0.0f;
    return v;
}

// D = A(16x4,f32) * B(4x16,f32) + C(16x16,f32)
__device__ __forceinline__ v8f wmma_f32(v2f a, v2f b, v8f c) {
    return __builtin_amdgcn_wmma_f32_16x16x4_f32(false, a, false, b, (short)0, c, false, false);
}

// ---------------------------------------------------------------------------
// Parallel input projection: xp[M, 4H] = X[M, IN] * Wih^T + (bih + bhh),
// M = B*T = 16384. One 16x16 tile per wave; full-chip WMMA GEMM.
// ---------------------------------------------------------------------------
template <int IN, int H>
__global__ void __launch_bounds__(256)
xp_gemm_kernel(const float* __restrict__ X,     // [16384, IN]
               const float* __restrict__ Wih,   // [4H, IN]
               const float* __restrict__ bih,
               const float* __restrict__ bhh,
               float* __restrict__ xp)          // [16384, 4H]
{
    constexpr int NT = (4 * H) / 16;
    const int tid  = threadIdx.x;
    const int lane = tid & 31;
    const int wave = tid >> 5;
    const int l15  = lane & 15;
    const int hsel = (lane >> 4) & 1;

    const int tile = blockIdx.x * 8 + wave;
    const int mt = tile / NT;
    const int nt = tile % NT;
    const int rowA = mt * 16 + l15;
    const int n    = nt * 16 + l15;

    v8f acc = vzero8();
    for (int k0 = 0; k0 < IN; k0 += 4) {
        v2f a = *(const v2f*)(X + (long long)rowA * IN + k0 + hsel * 2);
        v2f b = *(const v2f*)(Wih + (long long)n * IN + k0 + hsel * 2);
        acc = wmma_f32(a, b, acc);
    }
    const float bv = bih[n] + bhh[n];
#pragma unroll
    for (int r = 0; r < 8; ++r) {
        int m = mt * 16 + r + hsel * 8;
        xp[(long long)m * (4 * H) + n] = acc[r] + bv;
    }
}

// ---------------------------------------------------------------------------
// Persistent LSTM scan. One workgroup. Each wave owns TPW tiles of
// 16(batch) x 16(hidden); gate accumulators (i,f,g,o) = 4 x v8f per tile;
// cell state c = v8f per tile, resident in VGPRs across all T steps.
// USE_XP: accumulators init from precomputed xp (bias included), recurrent
// K-loop over H only. Otherwise the x*Wih^T GEMM is fused into the scan.
// ---------------------------------------------------------------------------
template <int IN, int H, int TPW, bool USE_XP>
__global__ void __launch_bounds__(1024)
lstm_scan_kernel(const float* __restrict__ x, long long xStrideB, long long xStrideT,
                 const float* __restrict__ Wih,   // [4H, IN]
                 const float* __restrict__ Whh,   // [4H, H]
                 const float* __restrict__ bih,   // [4H]
                 const float* __restrict__ bhh,   // [4H]
                 const float* __restrict__ xp4h,  // [B*T, 4H] or nullptr
                 float* __restrict__ hs_out,      // [B, T, H]
                 float* __restrict__ hT_out)      // [B, H] or nullptr
{
    constexpr int NJT = H / 16;
    __shared__ float h_lds[B_ * H];   // <= 64 KB (H<=128)

    const int tid  = threadIdx.x;
    const int lane = tid & 31;
    const int wave = tid >> 5;
    const int l15  = lane & 15;
    const int hsel = (lane >> 4) & 1;       // lanes 16..31 take K+2 / M+8

    for (int i = tid; i < B_ * H; i += blockDim.x) h_lds[i] = 0.0f;

    const int mb     = wave / (NJT / TPW);
    const int njbase = (wave % (NJT / TPW)) * TPW;
    const int rowA   = mb * 16 + l15;

    // Bias depends only on the output column (lane). In xp mode it is
    // already folded into xp, so use 0 (folded at compile time).
    float bias[TPW][4];
#pragma unroll
    for (int p = 0; p < TPW; ++p)
#pragma unroll
        for (int g = 0; g < 4; ++g) {
            int n = g * H + (njbase + p) * 16 + l15;
            bias[p][g] = USE_XP ? 0.0f : (bih[n] + bhh[n]);
        }

    v8f c[TPW];
#pragma unroll
    for (int p = 0; p < TPW; ++p) c[p] = vzero8();

    __syncthreads();

    for (int t = 0; t < T_; ++t) {
        v8f acc[TPW][4];

        if (USE_XP) {
            // Init accumulators from precomputed projection (C layout).
#pragma unroll
            for (int p = 0; p < TPW; ++p)
#pragma unroll
                for (int g = 0; g < 4; ++g) {
                    const int col = g * H + (njbase + p) * 16 + l15;
#pragma unroll
                    for (int r = 0; r < 8; ++r) {
                        const int bb = mb * 16 + r + hsel * 8;
                        acc[p][g][r] = xp4h[((long long)bb * T_ + t) * (4 * H) + col];
                    }
                }
            // Prefetch next timestep's xp rows (uniform guard, EXEC stays full).
            if (t + 1 < T_) {
                __builtin_prefetch(xp4h + ((long long)rowA * T_ + (t + 1)) * (4 * H), 0, 0);
            }
        } else {
#pragma unroll
            for (int p = 0; p < TPW; ++p)
#pragma unroll
                for (int g = 0; g < 4; ++g) acc[p][g] = vzero8();

            // ---- fused x_t * Wih^T ----
            const float* xrow = x + (long long)rowA * xStrideB + (long long)t * xStrideT;
            if (t + 1 < T_) {
                __builtin_prefetch(xrow + xStrideT, 0, 0);
            }
            for (int k0 = 0; k0 < IN; k0 += 4) {
                v2f a = *(const v2f*)(xrow + k0 + hsel * 2);
#pragma unroll
                for (int p = 0; p < TPW; ++p)
#pragma unroll
                    for (int g = 0; g < 4; ++g) {
                        int n = g * H + (njbase + p) * 16 + l15;
                        v2f b = *(const v2f*)(Wih + (long long)n * IN + k0 + hsel * 2);
                        acc[p][g] = wmma_f32(a, b, acc[p][g]);
                    }
            }
        }

        // ---- h_{t-1} * Whh^T (recurrent critical path) ----
        for (int k0 = 0; k0 < H; k0 += 4) {
            v2f a = *(const v2f*)(&h_lds[rowA * H + k0 + hsel * 2]);
#pragma unroll
            for (int p = 0; p < TPW; ++p)
#pragma unroll
                for (int g = 0; g < 4; ++g) {
                    int n = g * H + (njbase + p) * 16 + l15;
                    v2f b = *(const v2f*)(Whh + (long long)n * H + k0 + hsel * 2);
                    acc[p][g] = wmma_f32(a, b, acc[p][g]);
                }
        }
        __syncthreads();   // all h_lds reads done before overwrite

        // ---- gates + state update (all in registers) ----
#pragma unroll
        for (int p = 0; p < TPW; ++p) {
            const int j = (njbase + p) * 16 + l15;
#pragma unroll
            for (int r = 0; r < 8; ++r) {
                const int bb = mb * 16 + r + hsel * 8;   // batch row of this C element
                float zi = acc[p][0][r] + bias[p][0];
                float zf = acc[p][1][r] + bias[p][1];
                float zg = acc[p][2][r] + bias[p][2];
                float zo = acc[p][3][r] + bias[p][3];
                float cn = sigm(zf) * c[p][r] + sigm(zi) * tanhf(zg);
                float hv = sigm(zo) * tanhf(cn);
                c[p][r] = cn;
                h_lds[bb * H + j] = hv;
                hs_out[((long long)bb * T_ + t) * H + j] = hv;
                if (hT_out && t == T_ - 1) hT_out[bb * H + j] = hv;
            }
        }
        __syncthreads();   // h_lds valid for next step
    }
}

// ---------------------------------------------------------------------------
// Final projection: out[M,64] = A[M,128] * Wout^T + bout, M = B*T = 16384.
// ---------------------------------------------------------------------------
__global__ void __launch_bounds__(256)
out_gemm_kernel(const float* __restrict__ A,     // [16384, 128]
                const float* __restrict__ Wout,  // [64, 128]
                const float* __restrict__ bout,  // [64]
                float* __restrict__ out)         // [16384, 64]
{
    const int tid  = threadIdx.x;
    const int lane = tid & 31;
    const int wave = tid >> 5;
    const int l15  = lane & 15;
    const int hsel = (lane >> 4) & 1;

    const int tile = blockIdx.x * 8 + wave;   // 512 blocks * 8 waves = 4096
    const int mt = tile >> 2;
    const int nt = tile & 3;
    const int rowA = mt * 16 + l15;
    const int n    = nt * 16 + l15;

    v8f acc = vzero8();
    for (int k0 = 0; k0 < 128; k0 += 4) {
        v2f a = *(const v2f*)(A + (long long)rowA * 128 + k0 + hsel * 2);
        v2f b = *(const v2f*)(Wout + (long long)n * 128 + k0 + hsel * 2);
        acc = wmma_f32(a, b, acc);
    }
    const float bv = bout[n];
#pragma unroll
    for (int r = 0; r < 8; ++r) {
        int m = mt * 16 + r + hsel * 8;
        out[(long long)m * 64 + n] = acc[r] + bv;
    }
}

// ---------------------------------------------------------------------------
// x.mean(axis=1):  [B,T,D] -> [B,D]
// ---------------------------------------------------------------------------
__global__ void mean_kernel(const float* __restrict__ x, float* __restrict__ xmean)
{
    int i = blockIdx.x * blockDim.x + threadIdx.x;
    if (i >= B_ * D_) return;
    int b = i / D_, d = i % D_;
    float s = 0.0f;
    for (int t = 0; t < T_; ++t) s += x[((long long)b * T_ + t) * D_ + d];
    xmean[i] = s * (1.0f / T_);
}

// ---------------------------------------------------------------------------
// feat_attn: score_d = sum_a Wv[a]*tanh(Wa[a,0]*xv_d + u[b,a]),
//            u[b,a] = ba[a] + sum_j Wa[a,1+j]*h[b,j]; softmax over D.
// ---------------------------------------------------------------------------
__global__ void __launch_bounds__(64)
feat_attn_kernel(const float* __restrict__ xsrc, int rowsPerB,
                 const float* __restrict__ h,    // [B, 32]
                 const float* __restrict__ Wa,   // [32, 33]
                 const float* __restrict__ ba,   // [32]
                 const float* __restrict__ Wv,   // [32]
                 float* __restrict__ out)        // rows x D
{
    __shared__ float u[HA_];
    __shared__ float red[D_];
    const int row = blockIdx.x;
    const int d   = threadIdx.x;
    const int b   = row / rowsPerB;

    if (d < HA_) {
        float s = ba[d];
        for (int j = 0; j < HA_; ++j) s += Wa[d * 33 + 1 + j] * h[b * HA_ + j];
        u[d] = s;
    }
    __syncthreads();

    const float xv = xsrc[(long long)row * D_ + d];
    float sc = 0.0f;
    for (int a = 0; a < HA_; ++a) sc += Wv[a] * tanhf(Wa[a * 33] * xv + u[a]);

    red[d] = sc;
    __syncthreads();
    float m = -1e30f;
    for (int i = 0; i < D_; ++i) m = fmaxf(m, red[i]);
    __syncthreads();
    float e = __expf(sc - m);
    red[d] = e;
    __syncthreads();
    float s2 = 0.0f;
    for (int i = 0; i < D_; ++i) s2 += red[i];
    out[(long long)row * D_ + d] = e / s2;
}

// combined = x * attn_w[b,None,:] * attn_m
__global__ void combined_kernel(const float* __restrict__ x,
                                const float* __restrict__ aw,  // [B,D]
                                const float* __restrict__ am,  // [B,T,D]
                                float* __restrict__ comb)
{
    long long i = (long long)blockIdx.x * blockDim.x + threadIdx.x;
    if (i >= (long long)B_ * T_ * D_) return;
    int d = (int)(i % D_);
    int b = (int)(i / ((long long)T_ * D_));
    comb[i] = x[i] * aw[b * D_ + d] * am[i];
}

// ---------------------------------------------------------------------------
// Host side
// ---------------------------------------------------------------------------
struct P {
    const float *x;
    const float *e1Wih,*e1Whh,*e1bih,*e1bhh;
    const float *e2Wih,*e2Whh,*e2bih,*e2bhh;
    const float *btWih,*btWhh,*btbih,*btbhh;
    const float *d1Wih,*d1Whh,*d1bih,*d1bhh;
    const float *d2Wih,*d2Whh,*d2bih,*d2bhh;
    const float *Wout,*bout;
    const float *faWa,*faba,*faWv;
    const float *fatWa,*fatba,*fatWv;
};

extern "C" void kernel_launch(void* const* d_in, const int* in_sizes, int n_in,
                              void* d_out, int out_size, void* d_ws, size_t ws_size,
                              hipStream_t stream)
{
    P p;
    const float* const* in = (const float* const*)d_in;
    if (n_in >= 1 && in_sizes[0] == B_ * T_ * D_) {
        // Insertion order: x, enc1{Wih,Whh,bih,bhh}, enc2, bott, dec1, dec2,
        //                  Wout, bout, fa{Wa,ba,Wv}, fat{Wa,ba,Wv}
        p.x = in[0];
        p.e1Wih=in[1];  p.e1Whh=in[2];  p.e1bih=in[3];  p.e1bhh=in[4];
        p.e2Wih=in[5];  p.e2Whh=in[6];  p.e2bih=in[7];  p.e2bhh=in[8];
        p.btWih=in[9];  p.btWhh=in[10]; p.btbih=in[11]; p.btbhh=in[12];
        p.d1Wih=in[13]; p.d1Whh=in[14]; p.d1bih=in[15]; p.d1bhh=in[16];
        p.d2Wih=in[17]; p.d2Whh=in[18]; p.d2bih=in[19]; p.d2bhh=in[20];
        p.Wout=in[21];  p.bout=in[22];
        p.faWa=in[23];  p.faba=in[24];  p.faWv=in[25];
        p.fatWa=in[26]; p.fatba=in[27]; p.fatWv=in[28];
    } else {
        // jax tree_flatten (sorted keys): params{Wout, bott{Whh,Wih,bhh,bih},
        // bout, dec1, dec2, enc1, enc2, fa{Wa,Wv,ba}, fat{Wa,Wv,ba}}, x
        p.Wout=in[0];
        p.btWhh=in[1];  p.btWih=in[2];  p.btbhh=in[3];  p.btbih=in[4];
        p.bout=in[5];
        p.d1Whh=in[6];  p.d1Wih=in[7];  p.d1bhh=in[8];  p.d1bih=in[9];
        p.d2Whh=in[10]; p.d2Wih=in[11]; p.d2bhh=in[12]; p.d2bih=in[13];
        p.e1Whh=in[14]; p.e1Wih=in[15]; p.e1bhh=in[16]; p.e1bih=in[17];
        p.e2Whh=in[18]; p.e2Wih=in[19]; p.e2bhh=in[20]; p.e2bih=in[21];
        p.faWa=in[22];  p.faWv=in[23];  p.faba=in[24];
        p.fatWa=in[25]; p.fatWv=in[26]; p.fatba=in[27];
        p.x=in[28];
    }

    // Workspace layout (floats)
    float* ws    = (float*)d_ws;
    float* hsA   = ws;                    // [B,T,128] = 2,097,152
    float* hsB   = hsA + 2097152;         // [B,T,64]  = 1,048,576
    float* hsC   = hsB + 1048576;         // [B,T,32]  =   524,288
    float* comb  = hsC + 524288;          // [B,T,64]  = 1,048,576
    float* xmean = comb + 1048576;        // [B,D]
    float* h1    = xmean + 8192;          // [B,32]
    float* h2    = h1 + 4096;             // [B,32]
    float* xp    = h2 + 4096;             // [B*T, 512] = 8,388,608 (xp path only)

    const size_t base_floats = 4734976;               // everything up to xp
    const size_t xp_floats   = 8388608;               // 16384 x 512
    const bool use_xp = ws_size >= (base_floats + xp_floats) * sizeof(float);

    float* out_main = (float*)d_out;                 // [B,T,D]
    float* out_aw   = out_main + B_ * T_ * D_;       // [B,D]
    float* out_am   = out_aw + B_ * D_;              // [B,T,D]

    // ---- encode(x) ----
    if (use_xp) {
        xp_gemm_kernel<64,128><<<4096, 256, 0, stream>>>(p.x, p.e1Wih, p.e1bih, p.e1bhh, xp);
        lstm_scan_kernel<64,128,2,true><<<1, 1024, 0, stream>>>(p.x, (long long)T_*64, 64,
            p.e1Wih, p.e1Whh, p.e1bih, p.e1bhh, xp, hsA, nullptr);
        xp_gemm_kernel<128,64><<<2048, 256, 0, stream>>>(hsA, p.e2Wih, p.e2bih, p.e2bhh, xp);
        lstm_scan_kernel<128,64,1,true><<<1, 1024, 0, stream>>>(hsA, (long long)T_*128, 128,
            p.e2Wih, p.e2Whh, p.e2bih, p.e2bhh, xp, hsB, nullptr);
        xp_gemm_kernel<64,32><<<1024, 256, 0, stream>>>(hsB, p.btWih, p.btbih, p.btbhh, xp);
        lstm_scan_kernel<64,32,1,true><<<1, 512, 0, stream>>>(hsB, (long long)T_*64, 64,
            p.btWih, p.btWhh, p.btbih, p.btbhh, xp, hsC, h1);
    } else {
        lstm_scan_kernel<64,128,2,false><<<1, 1024, 0, stream>>>(p.x, (long long)T_*64, 64,
            p.e1Wih, p.e1Whh, p.e1bih, p.e1bhh, nullptr, hsA, nullptr);
        lstm_scan_kernel<128,64,1,false><<<1, 1024, 0, stream>>>(hsA, (long long)T_*128, 128,
            p.e2Wih, p.e2Whh, p.e2bih, p.e2bhh, nullptr, hsB, nullptr);
        lstm_scan_kernel<64,32,1,false><<<1, 512, 0, stream>>>(hsB, (long long)T_*64, 64,
            p.btWih, p.btWhh, p.btbih, p.btbhh, nullptr, hsC, h1);
    }

    // ---- attention ----
    mean_kernel<<<(B_*D_ + 255)/256, 256, 0, stream>>>(p.x, xmean);
    feat_attn_kernel<<<B_,      64, 0, stream>>>(xmean, 1,  h1, p.faWa,  p.faba,  p.faWv,  out_aw);
    feat_attn_kernel<<<B_*T_,   64, 0, stream>>>(p.x,   T_, h1, p.fatWa, p.fatba, p.fatWv, out_am);
    combined_kernel<<<(B_*T_*D_ + 255)/256, 256, 0, stream>>>(p.x, out_aw, out_am, comb);

    // ---- encode(combined) ----
    if (use_xp) {
        xp_gemm_kernel<64,128><<<4096, 256, 0, stream>>>(comb, p.e1Wih, p.e1bih, p.e1bhh, xp);
        lstm_scan_kernel<64,128,2,true><<<1, 1024, 0, stream>>>(comb, (long long)T_*64, 64,
            p.e1Wih, p.e1Whh, p.e1bih, p.e1bhh, xp, hsA, nullptr);
        xp_gemm_kernel<128,64><<<2048, 256, 0, stream>>>(hsA, p.e2Wih, p.e2bih, p.e2bhh, xp);
        lstm_scan_kernel<128,64,1,true><<<1, 1024, 0, stream>>>(hsA, (long long)T_*128, 128,
            p.e2Wih, p.e2Whh, p.e2bih, p.e2bhh, xp, hsB, nullptr);
        xp_gemm_kernel<64,32><<<1024, 256, 0, stream>>>(hsB, p.btWih, p.btbih, p.btbhh, xp);
        lstm_scan_kernel<64,32,1,true><<<1, 512, 0, stream>>>(hsB, (long long)T_*64, 64,
            p.btWih, p.btWhh, p.btbih, p.btbhh, xp, hsC, h2);
    } else {
        lstm_scan_kernel<64,128,2,false><<<1, 1024, 0, stream>>>(comb, (long long)T_*64, 64,
            p.e1Wih, p.e1Whh, p.e1bih, p.e1bhh, nullptr, hsA, nullptr);
        lstm_scan_kernel<128,64,1,false><<<1, 1024, 0, stream>>>(hsA, (long long)T_*128, 128,
            p.e2Wih, p.e2Whh, p.e2bih, p.e2bhh, nullptr, hsB, nullptr);
        lstm_scan_kernel<64,32,1,false><<<1, 512, 0, stream>>>(hsB, (long long)T_*64, 64,
            p.btWih, p.btWhh, p.btbih, p.btbhh, nullptr, hsC, h2);
    }

    // ---- decoder (dec1 input = h2 broadcast over t: strideT = 0; fused) ----
    lstm_scan_kernel<32,64,1,false><<<1, 1024, 0, stream>>>(h2, 32, 0,
        p.d1Wih, p.d1Whh, p.d1bih, p.d1bhh, nullptr, hsB, nullptr);
    if (use_xp) {
        xp_gemm_kernel<64,128><<<4096, 256, 0, stream>>>(hsB, p.d2Wih, p.d2bih, p.d2bhh, xp);
        lstm_scan_kernel<64,128,2,true><<<1, 1024, 0, stream>>>(hsB, (long long)T_*64, 64,
            p.d2Wih, p.d2Whh, p.d2bih, p.d2bhh, xp, hsA, nullptr);
    } else {
        lstm_scan_kernel<64,128,2,false><<<1, 1024, 0, stream>>>(hsB, (long long)T_*64, 64,
            p.d2Wih, p.d2Whh, p.d2bih, p.d2bhh, nullptr, hsA, nullptr);
    }

    // ---- output projection ----
    out_gemm_kernel<<<512, 256, 0, stream>>>(hsA, p.Wout, p.bout, out_main);
}